// BatchHardQuadrupletLoss_25486335934654
// MI455X (gfx1250) — compile-verified
//
#include <hip/hip_runtime.h>
#include <math.h>

#define B_N 96
#define D_K 512
#define MARGIN_TRIPLE 0.2f
#define MARGIN_INTER  0.1f

typedef __attribute__((ext_vector_type(2))) float v2f;
typedef __attribute__((ext_vector_type(8))) float v8f;

// ---------------------------------------------------------------------------
// Kernel 1: row squared norms  sq[i] = sum_k E[i][k]^2   (diag of Gram)
// ---------------------------------------------------------------------------
__global__ void row_norms_kernel(const float* __restrict__ E, float* __restrict__ sq) {
    const int row = blockIdx.x;        // 0..95
    const int t   = threadIdx.x;       // 0..127
    const float* r = E + row * D_K;
    float s = 0.0f;
    for (int k = t; k < D_K; k += 128) s += r[k] * r[k];
    __shared__ float red[128];
    red[t] = s;
    __syncthreads();
    for (int off = 64; off > 0; off >>= 1) {
        if (t < off) red[t] += red[t + off];
        __syncthreads();
    }
    if (t == 0) sq[row] = red[0];
}

// ---------------------------------------------------------------------------
// Kernel 2: Gram tile via V_WMMA_F32_16X16X4_F32, fused into distance matrix
//   d[i][j] = sqrt(max(sq_i - 2*G_ij + sq_j, 0)), exact-zero preserved.
// One wave32 per 16x16 output tile; 6x6 tiles; K stepped by 4 (128 WMMA ops).
// A-layout (16x4 f32): lane l -> row (l&15), K offset (l>>4)*2, 2 floats.
// B-layout (4x16 f32): identical per-lane pattern on the other row block.
// C/D layout: VGPR r -> M = r + (lane>>4)*8, N = lane&15.
// ---------------------------------------------------------------------------
__global__ void gram_dist_kernel(const float* __restrict__ E,
                                 const float* __restrict__ sq,
                                 float* __restrict__ dmat) {
    const int tile = blockIdx.x;           // 0..35
    const int mi = tile / 6;
    const int ni = tile % 6;
    const int lane = threadIdx.x;          // 0..31 (one full wave, EXEC all 1s)
    const int lm = lane & 15;
    const int kh = (lane >> 4) << 1;       // 0 or 2

    const float* pa = E + (mi * 16 + lm) * D_K + kh;
    const float* pb = E + (ni * 16 + lm) * D_K + kh;

    v8f c = {};
    #pragma unroll 4
    for (int k0 = 0; k0 < D_K; k0 += 4) {
        v2f a = *(const v2f*)(pa + k0);    // {K=k0+kh, K=k0+kh+1}
        v2f b = *(const v2f*)(pb + k0);
        c = __builtin_amdgcn_wmma_f32_16x16x4_f32(
                /*neg_a=*/false, a, /*neg_b=*/false, b,
                /*c_mod=*/(short)0, c, /*reuse_a=*/false, /*reuse_b=*/false);
    }

    const int col   = ni * 16 + lm;
    const float sqj = sq[col];
    const int rbase = mi * 16 + ((lane >> 4) << 3);
    #pragma unroll
    for (int r = 0; r < 8; ++r) {
        const int row = rbase + r;
        float v = sq[row] - 2.0f * c[r] + sqj;
        v = fmaxf(v, 0.0f);
        dmat[row * B_N + col] = (v == 0.0f) ? 0.0f : sqrtf(v);
    }
}

// ---------------------------------------------------------------------------
// Kernel 3: batch-hard mining per anchor row i.
//   hardest_pos / pos_ind : argmax of mask_ap * d   (first-occurrence ties)
//   hardest_neg / neg_ind : argmin of d + max_d*(1-mask_an)
//   triplet_i = relu(hp - hn + 0.2) accumulated into acc[0].
// ---------------------------------------------------------------------------
__global__ void mine_kernel(const float* __restrict__ dmat,
                            const int*   __restrict__ y,
                            int* __restrict__ pos_ind, int* __restrict__ neg_ind,
                            float* __restrict__ acc) {
    const int i = blockIdx.x;              // 0..95
    const int t = threadIdx.x;             // 0..127
    __shared__ float sv[128];
    __shared__ int   si[128];

    const int   yi  = y[i];
    const float dij = (t < B_N) ? dmat[i * B_N + t] : 0.0f;
    const int   yt  = (t < B_N) ? y[t] : -1;

    // --- pass 1: unmasked row max (max_d) ---
    sv[t] = (t < B_N) ? dij : -1.0f;
    __syncthreads();
    for (int off = 64; off > 0; off >>= 1) {
        if (t < off) sv[t] = fmaxf(sv[t], sv[t + off]);
        __syncthreads();
    }
    const float maxd = sv[0];
    __syncthreads();

    // --- pass 2: argmax of mask_ap * d (masked entries are exactly 0) ---
    const bool  ap  = (t < B_N) && (t != i) && (yt == yi);
    sv[t] = (t < B_N) ? (ap ? dij : 0.0f) : -1.0f;
    si[t] = (t < B_N) ? t : 0x7fffffff;
    __syncthreads();
    for (int off = 64; off > 0; off >>= 1) {
        if (t < off) {
            const float v2 = sv[t + off]; const int i2 = si[t + off];
            if (v2 > sv[t] || (v2 == sv[t] && i2 < si[t])) { sv[t] = v2; si[t] = i2; }
        }
        __syncthreads();
    }
    const float hp = sv[0];
    const int   pi = si[0];
    __syncthreads();

    // --- pass 3: argmin of an = d + maxd * (same-identity incl. diagonal) ---
    const bool an_ok = (t < B_N) && (yt != yi);
    sv[t] = (t < B_N) ? (dij + (an_ok ? 0.0f : maxd)) : 3.0e38f;
    si[t] = (t < B_N) ? t : 0x7fffffff;
    __syncthreads();
    for (int off = 64; off > 0; off >>= 1) {
        if (t < off) {
            const float v2 = sv[t + off]; const int i2 = si[t + off];
            if (v2 < sv[t] || (v2 == sv[t] && i2 < si[t])) { sv[t] = v2; si[t] = i2; }
        }
        __syncthreads();
    }
    const float hn = sv[0];
    const int   ni = si[0];

    if (t == 0) {
        pos_ind[i] = pi;
        neg_ind[i] = ni;
        atomicAdd(&acc[0], fmaxf(hp - hn + MARGIN_TRIPLE, 0.0f));
    }
}

// ---------------------------------------------------------------------------
// Kernel 4: quadruplet inter-class term.
//   sel[a,b,l] = relu(d[b,p_a] - d[n_a,l] + 0.1) * qmask(b, p_a, n_a, l)
//   inter[a,l] = max_b sel[a,b,l]   (>= 0, masked entries contribute 0)
//   acc[1] += sum_{a,l} inter[a,l]
// qmask: (b!=p) & (y_b==y_p) & (y_p!=y_n) & (y_n!=y_l) & (y_b!=y_l)
// ---------------------------------------------------------------------------
__global__ void inter_kernel(const float* __restrict__ dmat,
                             const int*   __restrict__ y,
                             const int* __restrict__ pos_ind,
                             const int* __restrict__ neg_ind,
                             float* __restrict__ acc) {
    const int a = blockIdx.x;              // 0..95
    const int l = threadIdx.x;             // 0..127 (l < 96 active)
    __shared__ int ysh[B_N];
    if (l < B_N) ysh[l] = y[l];
    __syncthreads();

    const int p  = pos_ind[a];
    const int n  = neg_ind[a];
    const int yp = ysh[p];
    const int yn = ysh[n];

    float v = 0.0f;
    if (l < B_N) {
        const int   yl  = ysh[l];
        const float dnl = dmat[n * B_N + l];
        if (yp != yn && yn != yl) {
            for (int b = 0; b < B_N; ++b) {
                if (b != p && ysh[b] == yp && ysh[b] != yl) {
                    const float val = fmaxf(dmat[b * B_N + p] - dnl + MARGIN_INTER, 0.0f);
                    v = fmaxf(v, val);
                }
            }
        }
    }
    __shared__ float red[128];
    red[l] = v;
    __syncthreads();
    for (int off = 64; off > 0; off >>= 1) {
        if (l < off) red[l] += red[l + off];
        __syncthreads();
    }
    if (l == 0) atomicAdd(&acc[1], red[0]);
}

// ---------------------------------------------------------------------------
// Init / finalize
//   mean(triplet (B,1) + inter (B,1,B)) over the (B,B,B) broadcast
//   = mean(triplet) + mean(inter)
// ---------------------------------------------------------------------------
__global__ void init_acc_kernel(float* acc) { acc[0] = 0.0f; acc[1] = 0.0f; }

__global__ void finalize_kernel(const float* __restrict__ acc, float* __restrict__ out) {
    out[0] = acc[0] / (float)B_N + acc[1] / ((float)B_N * (float)B_N);
}

// ---------------------------------------------------------------------------
extern "C" void kernel_launch(void* const* d_in, const int* in_sizes, int n_in,
                              void* d_out, int out_size, void* d_ws, size_t ws_size,
                              hipStream_t stream) {
    (void)in_sizes; (void)n_in; (void)out_size; (void)ws_size;

    const float* E = (const float*)d_in[0];   // (96, 512) fp32
    const int*   y = (const int*)d_in[1];     // (96,) identities
    float* out = (float*)d_out;               // scalar fp32

    // Workspace layout (floats):
    //   [0, 9216)      d matrix (96x96)
    //   [9216, 9312)   sq norms (96)
    //   then ints: pos_ind(96), neg_ind(96), then acc(2 floats)
    float* ws      = (float*)d_ws;
    float* dmat    = ws;
    float* sq      = ws + B_N * B_N;
    int*   posind  = (int*)(ws + B_N * B_N + B_N);
    int*   negind  = posind + B_N;
    float* acc     = (float*)(negind + B_N);

    init_acc_kernel<<<1, 1, 0, stream>>>(acc);
    row_norms_kernel<<<B_N, 128, 0, stream>>>(E, sq);
    gram_dist_kernel<<<36, 32, 0, stream>>>(E, sq, dmat);
    mine_kernel<<<B_N, 128, 0, stream>>>(dmat, y, posind, negind, acc);
    inter_kernel<<<B_N, 128, 0, stream>>>(dmat, y, posind, negind, acc);
    finalize_kernel<<<1, 1, 0, stream>>>(acc, out);
}